// NeuronSAT_73650099192328
// MI455X (gfx1250) — compile-verified
//
#include <hip/hip_runtime.h>

// Problem constants (fixed by the reference setup_inputs()).
#define NV_C    50000      // n_vars
#define V2_C    100000     // 2*n_vars (literals)
#define NC_C    200000     // n_clauses
#define NTOT_C  300000     // V2 + NC
#define D_C     128
#define E_C     800000
#define ROUNDS_C 16

#define LDS_ROW 132        // padded row stride (floats): conflict-free, 8B aligned
#define GEMM_LDS_BYTES ((128 * LDS_ROW + 4 * 16 * LDS_ROW) * 4)

typedef float v2f __attribute__((ext_vector_type(2)));
typedef float v8f __attribute__((ext_vector_type(8)));

// ---------------------------------------------------------------- init state
__global__ __launch_bounds__(256)
void init_k(float* __restrict__ h, float* __restrict__ c,
            const float* __restrict__ liw, const float* __restrict__ lib,
            const float* __restrict__ ciw, const float* __restrict__ cib) {
  size_t idx = (size_t)blockIdx.x * blockDim.x + threadIdx.x;
  if (idx >= (size_t)NTOT_C * D_C) return;
  int j = (int)(idx & (D_C - 1));
  size_t row = idx >> 7;
  h[idx] = (row < V2_C) ? (liw[j] + lib[j]) : (ciw[j] + cib[j]);
  c[idx] = 0.0f;
}

// ---------------------------------------------------------------- zero fill
__global__ __launch_bounds__(256)
void fill0_k(float* __restrict__ p, size_t n) {
  size_t i = (size_t)blockIdx.x * blockDim.x + threadIdx.x;
  if (i < n) p[i] = 0.0f;
}

// ---------------------------------------------------------------- WMMA GEMM
// C[M x N] (+)= A[M x 128] * W[N x ldw(K-slice of 128)]^T (+ bias) (ReLU?)
// Block = 4 waves; all waves share one 128-col W chunk staged in LDS;
// each wave stages its own 16x128 A tile in LDS, then the K loop is pure
// ds_load_b64 + v_wmma_f32_16x16x4_f32 (8 accumulators = 16x128 strip).
template <bool ACCUM, bool RELU, bool NEGFLIP>
__global__ __launch_bounds__(128)
void gemm_wmma_k(const float* __restrict__ A,
                 const float* __restrict__ W, int ldw,
                 const float* __restrict__ bias,
                 float* __restrict__ C, int ldc,
                 int Mtiles) {
  extern __shared__ float lds[];
  float* Wl = lds;                                   // 128 rows x LDS_ROW

  const int tid   = threadIdx.x;
  const int wave  = tid >> 5;
  const int lane  = tid & 31;
  const int mtile = blockIdx.x * 4 + wave;
  const bool active = (mtile < Mtiles);              // wave-uniform
  const int n0 = blockIdx.y * 128;

  // Stage W chunk [128 x 128] -> LDS, coalesced float4 (all threads).
#pragma unroll
  for (int i = 0; i < 32; ++i) {
    int e   = i * 128 + tid;                         // float4 id in [0,4096)
    int row = e >> 5;
    int c4  = (e & 31) << 2;
    float4 v = *(const float4*)(W + (size_t)(n0 + row) * ldw + c4);
    *(float4*)(Wl + row * LDS_ROW + c4) = v;
  }

  // Stage this wave's A tile [16 x 128] -> LDS, coalesced float4.
  float* Al = lds + 128 * LDS_ROW + wave * 16 * LDS_ROW;
  if (active) {
#pragma unroll
    for (int i = 0; i < 16; ++i) {
      int e   = i * 32 + lane;                       // float4 id in [0,512)
      int row = e >> 5;
      int c4  = (e & 31) << 2;
      int arow = mtile * 16 + row;
      if (NEGFLIP) arow = (arow < NV_C) ? (arow + NV_C) : (arow - NV_C);
      float4 v = *(const float4*)(A + (size_t)arow * D_C + c4);
      *(float4*)(Al + row * LDS_ROW + c4) = v;
    }
  }
  __syncthreads();                                   // all threads reach this
  if (!active) return;

  const int half = lane >> 4;
  const int l    = lane & 15;
  const float* Af = Al + l * LDS_ROW + 2 * half;     // A frag: row l, K pair
  const float* Bf = Wl + l * LDS_ROW + 2 * half;     // B frag: col l, K pair

  const v8f zero = {0.f, 0.f, 0.f, 0.f, 0.f, 0.f, 0.f, 0.f};
  v8f acc[8];
#pragma unroll
  for (int t = 0; t < 8; ++t) acc[t] = zero;

#pragma unroll 4
  for (int k = 0; k < D_C; k += 4) {
    v2f a = *(const v2f*)(Af + k);
#pragma unroll
    for (int t = 0; t < 8; ++t) {
      v2f b = *(const v2f*)(Bf + t * 16 * LDS_ROW + k);
      acc[t] = __builtin_amdgcn_wmma_f32_16x16x4_f32(
          false, a, false, b, (short)0, acc[t], false, false);
    }
  }

#pragma unroll
  for (int t = 0; t < 8; ++t) {
    const int ncol = n0 + t * 16 + l;
    const float bv = bias ? bias[ncol] : 0.0f;
#pragma unroll
    for (int r = 0; r < 8; ++r) {
      size_t off = (size_t)(mtile * 16 + r + 8 * half) * ldc + ncol;
      float v = acc[t][r] + bv;
      if (ACCUM) v += C[off];
      if (RELU)  v = fmaxf(v, 0.0f);
      C[off] = v;
    }
  }
}

// ---------------------------------------------------------------- edge scatter
// dst[idx_out[e], :] += src[idx_in[e], :]   (one wave per edge, float4 lanes)
__global__ __launch_bounds__(128)
void scatter_add_k(float* __restrict__ dst, const float* __restrict__ src,
                   const int* __restrict__ idx_out, const int* __restrict__ idx_in) {
  int e = blockIdx.x * 4 + (threadIdx.x >> 5);
  if (e >= E_C) return;
  int lane = threadIdx.x & 31;
  const float4 v = *(const float4*)(src + (size_t)idx_in[e] * D_C + lane * 4);
  float* d = dst + (size_t)idx_out[e] * D_C + lane * 4;
  atomicAdd(d + 0, v.x);
  atomicAdd(d + 1, v.y);
  atomicAdd(d + 2, v.z);
  atomicAdd(d + 3, v.w);
}

// ---------------------------------------------------------------- LSTM gates
// G row layout (512): [i | f | g | o]; c2 = sig(f)*c + sig(i)*tanh(g);
// h = sig(o)*tanh(c2)
__global__ __launch_bounds__(256)
void lstm_k(const float* __restrict__ G, float* __restrict__ h,
            float* __restrict__ c, int rows) {
  size_t idx = (size_t)blockIdx.x * blockDim.x + threadIdx.x;
  if (idx >= (size_t)rows * D_C) return;
  size_t row = idx >> 7;
  int j = (int)(idx & (D_C - 1));
  const float* g = G + row * 512 + j;
  float gi = g[0], gf = g[128], gg = g[256], go = g[384];
  float si = 1.0f / (1.0f + expf(-gi));
  float sf = 1.0f / (1.0f + expf(-gf));
  float so = 1.0f / (1.0f + expf(-go));
  float c2 = sf * c[idx] + si * tanhf(gg);
  h[idx] = so * tanhf(c2);
  c[idx] = c2;
}

// ---------------------------------------------------------------- host side
static void launch_gemm(const float* A, const float* W, int ldw, const float* bias,
                        float* C, int ldc, int M, int N,
                        bool accum, bool relu, bool negflip, hipStream_t s) {
  int Mtiles = M / 16;
  dim3 grid((unsigned)((Mtiles + 3) / 4), (unsigned)(N / 128), 1);
  dim3 blk(128, 1, 1);
  size_t shmem = GEMM_LDS_BYTES;
  if (relu)
    gemm_wmma_k<false, true, false><<<grid, blk, shmem, s>>>(A, W, ldw, bias, C, ldc, Mtiles);
  else if (negflip)
    gemm_wmma_k<true, false, true><<<grid, blk, shmem, s>>>(A, W, ldw, bias, C, ldc, Mtiles);
  else if (accum)
    gemm_wmma_k<true, false, false><<<grid, blk, shmem, s>>>(A, W, ldw, bias, C, ldc, Mtiles);
  else
    gemm_wmma_k<false, false, false><<<grid, blk, shmem, s>>>(A, W, ldw, bias, C, ldc, Mtiles);
}

extern "C" void kernel_launch(void* const* d_in, const int* in_sizes, int n_in,
                              void* d_out, int out_size, void* d_ws, size_t ws_size,
                              hipStream_t stream) {
  (void)in_sizes; (void)n_in; (void)out_size; (void)ws_size;

  const float* L_init_w = (const float*)d_in[0];
  const float* L_init_b = (const float*)d_in[1];
  const float* C_init_w = (const float*)d_in[2];
  const float* C_init_b = (const float*)d_in[3];
  const float* Lmsg_w   = (const float*)d_in[4];
  const float* Lmsg_b   = (const float*)d_in[5];
  const float* Cmsg_w   = (const float*)d_in[6];
  const float* Cmsg_b   = (const float*)d_in[7];
  const float* Cu_wih   = (const float*)d_in[8];
  const float* Cu_whh   = (const float*)d_in[9];
  const float* Cu_bih   = (const float*)d_in[10];
  const float* Cu_bhh   = (const float*)d_in[11];
  const float* Lu_wih   = (const float*)d_in[12];   // [512, 256]
  const float* Lu_whh   = (const float*)d_in[13];
  const float* Lu_bih   = (const float*)d_in[14];
  const float* Lu_bhh   = (const float*)d_in[15];
  const int*   edge_src = (const int*)d_in[16];
  const int*   edge_dst = (const int*)d_in[17];

  float* h    = (float*)d_out;                        // h state lives in d_out
  float* c    = (float*)d_ws;                         // NTOT*128
  float* bufA = c    + (size_t)NTOT_C * D_C;          // NC*128
  float* bufB = bufA + (size_t)NC_C * D_C;            // NC*128
  float* G    = bufB + (size_t)NC_C * D_C;            // NC*512 gate buffer

  float* h_cl = h + (size_t)V2_C * D_C;
  float* c_cl = c + (size_t)V2_C * D_C;

  {
    size_t n = (size_t)NTOT_C * D_C;
    init_k<<<dim3((unsigned)((n + 255) / 256)), dim3(256), 0, stream>>>(
        h, c, L_init_w, L_init_b, C_init_w, C_init_b);
  }

  for (int r = 0; r < ROUNDS_C; ++r) {
    // ---- literal -> clause: lm = MLP3(h_lit) ----
    launch_gemm(h,    Lmsg_w + 0 * 16384, 128, Lmsg_b + 0,   bufA, 128, V2_C, 128, false, true,  false, stream);
    launch_gemm(bufA, Lmsg_w + 1 * 16384, 128, Lmsg_b + 128, bufB, 128, V2_C, 128, false, true,  false, stream);
    launch_gemm(bufB, Lmsg_w + 2 * 16384, 128, Lmsg_b + 256, bufA, 128, V2_C, 128, false, false, false, stream);

    {   // c_msg = segment_sum(lm[edge_src] -> edge_dst)
      size_t n = (size_t)NC_C * D_C;
      fill0_k<<<dim3((unsigned)((n + 255) / 256)), dim3(256), 0, stream>>>(bufB, n);
    }
    scatter_add_k<<<dim3(E_C / 4), dim3(128), 0, stream>>>(bufB, bufA, edge_dst, edge_src);

    // ---- clause LSTM: G = c_msg*Wih^T + h_cl*Whh^T + bih + bhh ----
    launch_gemm(bufB, Cu_wih, 128, Cu_bih, G, 512, NC_C, 512, false, false, false, stream);
    launch_gemm(h_cl, Cu_whh, 128, Cu_bhh, G, 512, NC_C, 512, true,  false, false, stream);
    {
      size_t n = (size_t)NC_C * D_C;
      lstm_k<<<dim3((unsigned)((n + 255) / 256)), dim3(256), 0, stream>>>(G, h_cl, c_cl, NC_C);
    }

    // ---- clause -> literal: cm = MLP3(h_cl) ----
    launch_gemm(h_cl, Cmsg_w + 0 * 16384, 128, Cmsg_b + 0,   bufA, 128, NC_C, 128, false, true,  false, stream);
    launch_gemm(bufA, Cmsg_w + 1 * 16384, 128, Cmsg_b + 128, bufB, 128, NC_C, 128, false, true,  false, stream);
    launch_gemm(bufB, Cmsg_w + 2 * 16384, 128, Cmsg_b + 256, bufA, 128, NC_C, 128, false, false, false, stream);

    {   // l_msg = segment_sum(cm[edge_dst] -> edge_src)
      size_t n = (size_t)V2_C * D_C;
      fill0_k<<<dim3((unsigned)((n + 255) / 256)), dim3(256), 0, stream>>>(bufB, n);
    }
    scatter_add_k<<<dim3(E_C / 4), dim3(128), 0, stream>>>(bufB, bufA, edge_src, edge_dst);

    // ---- literal LSTM: x = [l_msg | h_neg]; G = x*Wih^T + h_lit*Whh^T + b ----
    launch_gemm(bufB, Lu_wih,       256, Lu_bih,  G, 512, V2_C, 512, false, false, false, stream);
    launch_gemm(h,    Lu_wih + 128, 256, nullptr, G, 512, V2_C, 512, true,  false, true,  stream);
    launch_gemm(h,    Lu_whh,       128, Lu_bhh,  G, 512, V2_C, 512, true,  false, false, stream);
    {
      size_t n = (size_t)V2_C * D_C;
      lstm_k<<<dim3((unsigned)((n + 255) / 256)), dim3(256), 0, stream>>>(G, h, c, V2_C);
    }
  }
}